// SAGE_mol_23802708754782
// MI455X (gfx1250) — compile-verified
//
#include <hip/hip_runtime.h>
#include <hip/hip_bf16.h>
#include <math.h>

// ---------------------------------------------------------------------------
// Problem constants (from the reference)
// ---------------------------------------------------------------------------
#define Gg   128
#define Nn   1024
#define Hc   128          // F_IN == H == 128
#define DEG  8
#define K1c  820
#define K2c  410
#define NTOT (Gg * Nn)    // 131072 nodes, layer 1
#define N1   (Gg * K1c)   // 104960 nodes, layer 2
#define N2   (Gg * K2c)   // 52480 nodes after pool2

typedef __attribute__((ext_vector_type(2))) float v2f;
typedef __attribute__((ext_vector_type(8))) float v8f;

// Native fp32 global atomic add (GLOBAL_ATOMIC_ADD_F32 on gfx1250)
__device__ __forceinline__ void atomAddF32(float* p, float v) {
    (void)__hip_atomic_fetch_add(p, v, __ATOMIC_RELAXED, __HIP_MEMORY_SCOPE_AGENT);
}

// ---------------------------------------------------------------------------
// 1. Edge scatter: agg[dst] += x[src]; cnt[dst] += 1   (optionally relabeled)
//    One wave per edge: 32 lanes x 4 contiguous floats.
// ---------------------------------------------------------------------------
__global__ __launch_bounds__(256)
void scatter_edges(const int* __restrict__ src, const int* __restrict__ dst,
                   const float* __restrict__ xin, const int* __restrict__ new_id,
                   float* __restrict__ agg, float* __restrict__ cnt, int E)
{
    const int lane = threadIdx.x & 31;
    const int wv   = threadIdx.x >> 5;
    for (int e = blockIdx.x * 8 + wv; e < E; e += gridDim.x * 8) {
        int s = src[e], d = dst[e];
        if (new_id) {
            s = new_id[s];
            d = new_id[d];
            if ((s | d) < 0) continue;   // edge dropped by pooling
        }
        const float4 v = *(const float4*)(xin + (size_t)s * Hc + lane * 4);
        float* ad = agg + (size_t)d * Hc + lane * 4;
        atomAddF32(ad + 0, v.x);
        atomAddF32(ad + 1, v.y);
        atomAddF32(ad + 2, v.z);
        atomAddF32(ad + 3, v.w);
        if (lane == 0) atomAddF32(cnt + d, 1.0f);
    }
}

// ---------------------------------------------------------------------------
// 2. Fused SAGE GEMM:  h = relu( (agg/max(cnt,1)) @ Wl + bl + x @ Wr )
//    fp32 WMMA 16x16x4. Block = 8 waves; wave w owns output col tile w.
//    Per-wave: Wl/Wr col tiles (128x16 each) preloaded to registers,
//    grid-stride loop over 16-row tiles, K-loop of 32 steps, 2 WMMA each.
// ---------------------------------------------------------------------------
__global__ __launch_bounds__(256)
void sage_gemm(const float* __restrict__ agg, const float* __restrict__ cnt,
               const float* __restrict__ xin,
               const float* __restrict__ Wl, const float* __restrict__ bl,
               const float* __restrict__ Wr,
               float* __restrict__ hout, int nRows)
{
    const int lane = threadIdx.x & 31;
    const int wave = threadIdx.x >> 5;   // 0..7 -> output column tile
    const int n0   = wave * 16;
    const int m    = lane & 15;          // row-in-tile (A) / col-in-tile (B,C)
    const int hi   = lane >> 4;          // half-wave select (K offset of 2 / M offset of 8)

    // Preload B tiles. 16x16x4 B layout (4xK rows striped across lanes):
    // VGPR0 = K = 4k+2*hi, VGPR1 = K = 4k+2*hi+1, column = n0+m.
    v2f wl[32], wr[32];
#pragma unroll
    for (int k = 0; k < 32; ++k) {
        const int kk = 4 * k + 2 * hi;
        wl[k].x = Wl[(size_t)kk * Hc + n0 + m];
        wl[k].y = Wl[(size_t)(kk + 1) * Hc + n0 + m];
        wr[k].x = Wr[(size_t)kk * Hc + n0 + m];
        wr[k].y = Wr[(size_t)(kk + 1) * Hc + n0 + m];
    }
    const float bias   = bl[n0 + m];
    const int   nTiles = nRows >> 4;

    for (int rt = blockIdx.x; rt < nTiles; rt += gridDim.x) {
        const int   row  = (rt << 4) + m;                 // A row for this lane
        const float sc   = 1.0f / fmaxf(cnt[row], 1.0f);  // mean-aggregation scale
        const float* aggRow = agg + (size_t)row * Hc;
        const float* xRow   = xin + (size_t)row * Hc;
        v8f c = {};
#pragma unroll
        for (int k = 0; k < 32; ++k) {
            const int kk = 4 * k + 2 * hi;
            v2f a;  a.x  = aggRow[kk] * sc;  a.y  = aggRow[kk + 1] * sc;
            c = __builtin_amdgcn_wmma_f32_16x16x4_f32(false, a,  false, wl[k],
                                                      (short)0, c, false, false);
            v2f ax; ax.x = xRow[kk];         ax.y = xRow[kk + 1];
            c = __builtin_amdgcn_wmma_f32_16x16x4_f32(false, ax, false, wr[k],
                                                      (short)0, c, false, false);
        }
        // C layout: VGPR v -> M = v + 8*hi, N = n0 + m
#pragma unroll
        for (int v = 0; v < 8; ++v) {
            const int rowOut = (rt << 4) + v + 8 * hi;
            hout[(size_t)rowOut * Hc + n0 + m] = fmaxf(c[v] + bias, 0.0f);
        }
    }
}

// ---------------------------------------------------------------------------
// 3. Pool scoring: score[i] = tanh( dot(h[i], w) / ||w|| )
// ---------------------------------------------------------------------------
__global__ __launch_bounds__(128)
void rnorm_kernel(const float* __restrict__ w, float* __restrict__ out)
{
    __shared__ float red[128];
    const int t = threadIdx.x;
    const float v = w[t];
    red[t] = v * v;
    __syncthreads();
    for (int s = 64; s > 0; s >>= 1) {
        if (t < s) red[t] += red[t + s];
        __syncthreads();
    }
    if (t == 0) out[0] = 1.0f / sqrtf(red[0]);
}

__global__ __launch_bounds__(256)
void score_kernel(const float* __restrict__ h, const float* __restrict__ w,
                  const float* __restrict__ rnorm, float* __restrict__ score,
                  int nNodes)
{
    const int lane = threadIdx.x & 31;
    const int node = blockIdx.x * 8 + (threadIdx.x >> 5);
    if (node >= nNodes) return;
    const float4 hv = *(const float4*)(h + (size_t)node * Hc + lane * 4);
    const float4 wv = *(const float4*)(w + lane * 4);
    float s = hv.x * wv.x + hv.y * wv.y + hv.z * wv.z + hv.w * wv.w;
    for (int off = 16; off; off >>= 1) s += __shfl_xor(s, off, 32);
    if (lane == 0) score[node] = tanhf(s * rnorm[0]);
}

// ---------------------------------------------------------------------------
// 4. Per-graph top-k: bitonic sort (descending) of 1024 (score, idx) in LDS.
// ---------------------------------------------------------------------------
__global__ __launch_bounds__(1024)
void topk_sort(const float* __restrict__ score, int n_per, int k,
               int* __restrict__ perm, float* __restrict__ vals)
{
    __shared__ float sv[1024];
    __shared__ int   si[1024];
    const int g = blockIdx.x, t = threadIdx.x;
    sv[t] = (t < n_per) ? score[(size_t)g * n_per + t] : -3.0e38f;
    si[t] = t;
    __syncthreads();
    for (int kk = 2; kk <= 1024; kk <<= 1) {
        for (int j = kk >> 1; j > 0; j >>= 1) {
            const int ixj = t ^ j;
            if (ixj > t) {
                const bool up = ((t & kk) == 0);
                const float a = sv[t], b = sv[ixj];
                if ((a < b) == up) {           // descending sort
                    sv[t] = b;  sv[ixj] = a;
                    const int ia = si[t];
                    si[t] = si[ixj];  si[ixj] = ia;
                }
            }
            __syncthreads();
        }
    }
    if (t < k) {
        perm[(size_t)g * k + t] = g * n_per + si[t];   // global old node id
        vals[(size_t)g * k + t] = sv[t];               // tanh score (gate)
    }
}

// out[r] = h[perm[r]] * vals[r]
__global__ __launch_bounds__(256)
void gather_gate(const float* __restrict__ h, const int* __restrict__ perm,
                 const float* __restrict__ vals, float* __restrict__ out, int nrows)
{
    const int r = blockIdx.x * 2 + (threadIdx.x >> 7);
    const int f = threadIdx.x & 127;
    if (r < nrows)
        out[(size_t)r * Hc + f] = h[(size_t)perm[r] * Hc + f] * vals[r];
}

// new_id[perm[i]] = i
__global__ __launch_bounds__(256)
void relabel_kernel(const int* __restrict__ perm, int* __restrict__ new_id, int n)
{
    const int i = blockIdx.x * 256 + threadIdx.x;
    if (i < n) new_id[perm[i]] = i;
}

// ---------------------------------------------------------------------------
// 5. Readout: [max ; sum] over each graph's k rows -> xr[g][0:256]
// ---------------------------------------------------------------------------
__global__ __launch_bounds__(128)
void readout_kernel(const float* __restrict__ hp, int k, float* __restrict__ xr)
{
    const int g = blockIdx.x, f = threadIdx.x;
    const float* base = hp + (size_t)g * k * Hc + f;
    float mx = -3.0e38f, sm = 0.0f;
    for (int j = 0; j < k; ++j) {
        const float v = base[(size_t)j * Hc];
        mx = fmaxf(mx, v);
        sm += v;
    }
    xr[(size_t)g * 256 + f]       = mx;
    xr[(size_t)g * 256 + 128 + f] = sm;
}

// ---------------------------------------------------------------------------
// 6. MLP head
// ---------------------------------------------------------------------------
__global__ __launch_bounds__(256)
void head1_kernel(const float* __restrict__ x1, const float* __restrict__ x2,
                  const float* __restrict__ Wp1, const float* __restrict__ bp1,
                  float* __restrict__ z)
{
    const int idx = blockIdx.x * 256 + threadIdx.x;   // g*1024 + j
    const int g = idx >> 10, j = idx & 1023;
    float acc = bp1[j];
#pragma unroll 8
    for (int k = 0; k < 256; ++k)
        acc += (x1[(size_t)g * 256 + k] + x2[(size_t)g * 256 + k]) * Wp1[(size_t)k * 1024 + j];
    z[idx] = fmaxf(acc, 0.0f);
}

__global__ __launch_bounds__(128)
void head2_kernel(const float* __restrict__ z, const float* __restrict__ Wp2,
                  const float* __restrict__ bp2, float* __restrict__ out)
{
    __shared__ float r0[128], r1[128];
    const int g = blockIdx.x, t = threadIdx.x;
    float p0 = 0.0f, p1 = 0.0f;
    for (int k = t; k < 1024; k += 128) {
        const float v = z[(size_t)g * 1024 + k];
        p0 += v * Wp2[k * 2];
        p1 += v * Wp2[k * 2 + 1];
    }
    r0[t] = p0; r1[t] = p1;
    __syncthreads();
    for (int s = 64; s > 0; s >>= 1) {
        if (t < s) { r0[t] += r0[t + s]; r1[t] += r1[t + s]; }
        __syncthreads();
    }
    if (t == 0) {
        const float l0 = r0[0] + bp2[0], l1 = r1[0] + bp2[1];
        out[g * 2]     = l0;
        out[g * 2 + 1] = l1;
        const float mx  = fmaxf(l0, l1);
        const float lse = mx + logf(expf(l0 - mx) + expf(l1 - mx));
        out[2 * Gg + g * 2]     = l0 - lse;
        out[2 * Gg + g * 2 + 1] = l1 - lse;
    }
}

// ---------------------------------------------------------------------------
// Host driver
// ---------------------------------------------------------------------------
extern "C" void kernel_launch(void* const* d_in, const int* in_sizes, int n_in,
                              void* d_out, int out_size, void* d_ws, size_t ws_size,
                              hipStream_t stream)
{
    const float* x   = (const float*)d_in[0];
    const int*   ei  = (const int*)d_in[1];   // [2, E]
    const float* W1l = (const float*)d_in[4];
    const float* b1l = (const float*)d_in[5];
    const float* W1r = (const float*)d_in[6];
    const float* p1w = (const float*)d_in[7];
    const float* W2l = (const float*)d_in[8];
    const float* b2l = (const float*)d_in[9];
    const float* W2r = (const float*)d_in[10];
    const float* p2w = (const float*)d_in[11];
    const float* Wp1 = (const float*)d_in[12];
    const float* bp1 = (const float*)d_in[13];
    const float* Wp2 = (const float*)d_in[14];
    const float* bp2 = (const float*)d_in[15];
    float* out = (float*)d_out;

    const int E = in_sizes[1] / 2;
    const int* src = ei;
    const int* dst = ei + E;

    // ---- workspace carving (floats/ints, all even offsets) ----
    float* ws   = (float*)d_ws;
    float* aggA = ws;                                 // NTOT*Hc (agg1, later agg for reuse never needed; h2 below)
    float* bufB = aggA + (size_t)NTOT * Hc;           // NTOT*Hc (h1, later h2)
    float* hp   = bufB + (size_t)NTOT * Hc;           // N1*Hc
    float* hp2  = hp   + (size_t)N1 * Hc;             // N2*Hc
    float* cnt  = hp2  + (size_t)N2 * Hc;             // NTOT (reused for cnt2)
    float* scr  = cnt  + NTOT;                        // NTOT (reused)
    float* v1   = scr  + NTOT;                        // N1
    float* v2   = v1   + N1;                          // N2
    int*   perm1 = (int*)(v2 + N2);                   // N1
    int*   perm2 = perm1 + N1;                        // N2
    int*   nid   = perm2 + N2;                        // NTOT
    float* x1    = (float*)(nid + NTOT);              // Gg*256
    float* x2    = x1 + Gg * 256;                     // Gg*256
    float* z     = x2 + Gg * 256;                     // Gg*1024
    float* rn    = z  + Gg * 1024;                    // 2

    // ================= conv1 =================
    hipMemsetAsync(aggA, 0, (size_t)NTOT * Hc * sizeof(float), stream);
    hipMemsetAsync(cnt,  0, (size_t)NTOT * sizeof(float), stream);
    scatter_edges<<<8192, 256, 0, stream>>>(src, dst, x, nullptr, aggA, cnt, E);
    sage_gemm<<<2048, 256, 0, stream>>>(aggA, cnt, x, W1l, b1l, W1r, bufB, NTOT);

    // ================= pool1 =================
    rnorm_kernel<<<1, 128, 0, stream>>>(p1w, rn);
    score_kernel<<<NTOT / 8, 256, 0, stream>>>(bufB, p1w, rn, scr, NTOT);
    topk_sort<<<Gg, 1024, 0, stream>>>(scr, Nn, K1c, perm1, v1);
    gather_gate<<<N1 / 2, 256, 0, stream>>>(bufB, perm1, v1, hp, N1);
    readout_kernel<<<Gg, 128, 0, stream>>>(hp, K1c, x1);

    // ================= filter_adj + conv2 =================
    hipMemsetAsync(nid, 0xFF, (size_t)NTOT * sizeof(int), stream);
    relabel_kernel<<<(N1 + 255) / 256, 256, 0, stream>>>(perm1, nid, N1);
    hipMemsetAsync(aggA, 0, (size_t)N1 * Hc * sizeof(float), stream);
    hipMemsetAsync(cnt,  0, (size_t)N1 * sizeof(float), stream);
    scatter_edges<<<8192, 256, 0, stream>>>(src, dst, hp, nid, aggA, cnt, E);
    sage_gemm<<<2048, 256, 0, stream>>>(aggA, cnt, hp, W2l, b2l, W2r, bufB, N1);

    // ================= pool2 =================
    rnorm_kernel<<<1, 128, 0, stream>>>(p2w, rn + 1);
    score_kernel<<<N1 / 8, 256, 0, stream>>>(bufB, p2w, rn + 1, scr, N1);
    topk_sort<<<Gg, 1024, 0, stream>>>(scr, K1c, K2c, perm2, v2);
    gather_gate<<<N2 / 2, 256, 0, stream>>>(bufB, perm2, v2, hp2, N2);
    readout_kernel<<<Gg, 128, 0, stream>>>(hp2, K2c, x2);

    // ================= MLP head =================
    head1_kernel<<<(Gg * 1024) / 256, 256, 0, stream>>>(x1, x2, Wp1, bp1, z);
    head2_kernel<<<Gg, 128, 0, stream>>>(z, Wp2, bp2, out);
}